// ImprovedDiffusionLayer_18322330485154
// MI455X (gfx1250) — compile-verified
//
#include <hip/hip_runtime.h>
#include <math.h>
#include <stdint.h>

#define IMG_W 512
#define IMG_H 512
#define IMG_B 64
#define KDT   0.01f

// Tile geometry: 128x32 core, x-halo padded to 4 (keeps 16B alignment of all
// LDS vector accesses), y-halo 2 -> 136x36 region = 19,584 B LDS.
#define TW 128
#define TH 32
#define HX 4
#define RW (TW + 2 * HX)   // 136
#define RH (TH + 4)        // 36

typedef __attribute__((ext_vector_type(4))) unsigned int u32x4;
typedef __attribute__((ext_vector_type(4))) int          i32x4;
typedef __attribute__((ext_vector_type(8))) int          i32x8;

#if defined(__has_builtin)
# if __has_builtin(__builtin_amdgcn_tensor_load_to_lds)
#  define HAVE_TDM 1
# endif
#endif
#ifndef HAVE_TDM
# define HAVE_TDM 0
#endif

// ---------------------------------------------------------------------------
// Reference-faithful stencil math
// ---------------------------------------------------------------------------

// Derivative of u at index `pos` (h = 1): one-sided at boundaries, central
// elsewhere. um = u[pos-1], u0 = u[pos], up = u[pos+1].
__device__ __forceinline__ float grad_at(float um, float u0, float up, int pos, int n) {
    if (pos == 0)     return up - u0;
    if (pos == n - 1) return u0 - um;
    return 0.5f * (up - um);
}

// d/dx of (cfl * d/dx u) at x given the 5-point window u[x-2..x+2].
// Same grad -> flux -> diff-of-flux rounding order as the reference.
__device__ __forceinline__ float div_flux_1d(float m2, float m1, float c0,
                                             float p1, float p2,
                                             int x, int n, float cfl) {
    float fm = cfl * grad_at(m2, m1, c0, x - 1, n);  // flux at x-1 (unused if x==0)
    float fp = cfl * grad_at(c0, p1, p2, x + 1, n);  // flux at x+1 (unused if x==n-1)
    if (x == 0) {
        float f0 = cfl * (p1 - c0);
        return fp - f0;
    }
    if (x == n - 1) {
        float f0 = cfl * (c0 - m1);
        return f0 - fm;
    }
    return 0.5f * (fp - fm);
}

// ---------------------------------------------------------------------------
// TDM 2D tile load: one DMA descriptor per workgroup (issued by wave 0).
// D# layout per CDNA5 ISA (2D tensor, 4B data).
// ---------------------------------------------------------------------------
__device__ __forceinline__ void tdm_load_tile_2d(unsigned lds_byte_addr,
                                                 const void* gptr) {
    const unsigned long long ga = (unsigned long long)(uintptr_t)gptr;
    u32x4 g0;
    g0.x = 1u;                                   // count=1, user descriptor, no gather
    g0.y = lds_byte_addr;                        // D#.lds_addr (bytes)
    g0.z = (unsigned)(ga & 0xffffffffu);         // global_addr[31:0]
    g0.w = (unsigned)((ga >> 32) & 0x1ffffffu)   // global_addr[56:32]
         | (2u << 30);                           // type = 2 ("image")

    i32x8 g1;
    g1[0] = (int)(2u << 16);                     // workgroup_mask=0, data_size=2 (4B)
    g1[1] = (int)(512u << 16);                   // tensor_dim0[15:0] = 512
    g1[2] = (int)(512u << 16);                   // tensor_dim0[31:16]=0 | tensor_dim1[15:0]=512
    g1[3] = (int)((unsigned)RW << 16);           // tensor_dim1[31:16]=0 | tile_dim0=136
    g1[4] = RH;                                  // tile_dim1=36 | tile_dim2=0
    g1[5] = IMG_W;                               // tensor_dim0_stride[31:0] = 512 elems
    g1[6] = 0;                                   // stride hi | tensor_dim1_stride lo
    g1[7] = 0;                                   // tensor_dim1_stride hi (unused, 2D)

    const i32x4 gz = {0, 0, 0, 0};
#if HAVE_TDM
# if __clang_major__ >= 23
    const i32x8 gz8 = {0, 0, 0, 0, 0, 0, 0, 0};
    __builtin_amdgcn_tensor_load_to_lds(g0, g1, gz, gz, gz8, 0);
# else
    __builtin_amdgcn_tensor_load_to_lds(g0, g1, gz, gz, 0);
# endif
    __builtin_amdgcn_s_wait_tensorcnt(0);
#else
    (void)g0; (void)g1; (void)gz;
#endif
}

// ---------------------------------------------------------------------------
// One diffusion step per kernel; tile staged in LDS via TDM, all LDS reads
// are 16B-aligned ds_load_b128. BLEND=true fuses the final sigmoid mix.
// ---------------------------------------------------------------------------
template <bool BLEND>
__global__ __launch_bounds__(256) void diffusion_step_tdm_kernel(
    const float* __restrict__ src, const float* __restrict__ orig,
    float* __restrict__ dst,
    const float* __restrict__ g_alpha, const float* __restrict__ g_beta,
    const float* __restrict__ g_ax, const float* __restrict__ g_ay,
    const float* __restrict__ g_es) {
    __shared__ float tile[RH * RW];

    const int cx0 = blockIdx.x * TW;
    const int cy0 = blockIdx.y * TH;
    const size_t plane = (size_t)blockIdx.z * (size_t)(IMG_W * IMG_H);
    // Clamp the halo'd read window fully in-bounds. All x clamps are
    // multiples of 4, preserving float4 alignment of core columns; clamped
    // cells are only consumed by positions whose boundary formula ignores
    // them (one-sided differences at the image edges).
    const int rx0 = min(max(cx0 - HX, 0), IMG_W - RW);
    const int ry0 = min(max(cy0 - 2, 0), IMG_H - RH);
    const float* img = src + plane;

    const int flat = threadIdx.x;

#if HAVE_TDM
    if ((flat >> 5) == 0) {  // wave 0 only: one DMA per workgroup
        tdm_load_tile_2d((unsigned)(uintptr_t)tile,
                         img + (size_t)ry0 * IMG_W + rx0);
    }
#else
    for (int i = flat; i < RW * RH; i += 256) {
        const int ly = i / RW, lx = i - ly * RW;
        tile[i] = img[(size_t)(ry0 + ly) * IMG_W + (rx0 + lx)];
    }
#endif
    __syncthreads();

    const float cxc = g_alpha[0] * g_ax[0];
    const float cyc = g_beta[0]  * g_ay[0];

    // 256 threads: 32 x-groups of 4 points x 8 rows, 4 row-iterations = 128x32.
    const int tx  = flat & 31;
    const int ty0 = flat >> 5;
    const int gx4 = cx0 + tx * 4;
    const int lx4 = gx4 - rx0;               // multiple of 4 by construction

    const int bx0 = max(lx4 - 4, 0);         // clamped, stays multiple of 4
    const int bx2 = min(lx4 + 4, RW - 4);

#pragma unroll
    for (int r = 0; r < 4; ++r) {
        const int ty = ty0 + 8 * r;
        const int gy = cy0 + ty;
        const int ly = gy - ry0;

        const int lym2 = max(ly - 2, 0), lym1 = max(ly - 1, 0);
        const int lyp1 = min(ly + 1, RH - 1), lyp2 = min(ly + 2, RH - 1);

        // All 16B-aligned -> ds_load_b128
        const float4 B0  = *(const float4*)(tile + ly   * RW + bx0);
        const float4 B1  = *(const float4*)(tile + ly   * RW + lx4);
        const float4 B2  = *(const float4*)(tile + ly   * RW + bx2);
        const float4 Ym2 = *(const float4*)(tile + lym2 * RW + lx4);
        const float4 Ym1 = *(const float4*)(tile + lym1 * RW + lx4);
        const float4 Yp1 = *(const float4*)(tile + lyp1 * RW + lx4);
        const float4 Yp2 = *(const float4*)(tile + lyp2 * RW + lx4);

        // x window u[gx4-2 .. gx4+5]
        const float w[8] = {B0.z, B0.w, B1.x, B1.y, B1.z, B1.w, B2.x, B2.y};
        const float* ym2 = (const float*)&Ym2;
        const float* ym1 = (const float*)&Ym1;
        const float* yp1 = (const float*)&Yp1;
        const float* yp2 = (const float*)&Yp2;

        float o[4];
#pragma unroll
        for (int j = 0; j < 4; ++j) {
            const int   gx = gx4 + j;
            const float c0 = w[j + 2];
            const float dxv = div_flux_1d(w[j], w[j + 1], c0, w[j + 3], w[j + 4],
                                          gx, IMG_W, cxc);
            const float dyv = div_flux_1d(ym2[j], ym1[j], c0, yp1[j], yp2[j],
                                          gy, IMG_H, cyc);
            o[j] = c0 + KDT * (dxv + dyv);
        }

        float4 o4 = make_float4(o[0], o[1], o[2], o[3]);
        const size_t off = plane + (size_t)gy * IMG_W + gx4;
        if (BLEND) {
            const float mix = 1.0f / (1.0f + expf(-g_es[0]));
            const float4 u0 = *(const float4*)(orig + off);
            o4.x = mix * o4.x + (1.0f - mix) * u0.x;
            o4.y = mix * o4.y + (1.0f - mix) * u0.y;
            o4.z = mix * o4.z + (1.0f - mix) * u0.z;
            o4.w = mix * o4.w + (1.0f - mix) * u0.w;
        }
        *(float4*)(dst + off) = o4;
    }
}

// ---------------------------------------------------------------------------
extern "C" void kernel_launch(void* const* d_in, const int* in_sizes, int n_in,
                              void* d_out, int out_size, void* d_ws, size_t ws_size,
                              hipStream_t stream) {
    const float* u     = (const float*)d_in[0];  // [64,1,512,512] f32
    const float* alpha = (const float*)d_in[1];
    const float* beta  = (const float*)d_in[2];
    const float* ax    = (const float*)d_in[3];
    const float* ay    = (const float*)d_in[4];
    const float* es    = (const float*)d_in[5];

    float* out = (float*)d_out;
    float* ws  = (float*)d_ws;   // one 64 MiB ping buffer

    dim3 block(256, 1, 1);                        // 8 wave32 waves
    dim3 grid(IMG_W / TW, IMG_H / TH, IMG_B);     // (4, 16, 64)

    // Steps 1..9 ping-pong: odd steps -> ws, even steps -> out.
    // All intermediates stay resident in the 192 MB L2.
    const float* s = u;
    for (int k = 1; k <= 9; ++k) {
        float* d = (k & 1) ? ws : out;
        diffusion_step_tdm_kernel<false><<<grid, block, 0, stream>>>(
            s, u, d, alpha, beta, ax, ay, es);
        s = d;  // after step 9, s == ws
    }
    // Step 10 fused with the sigmoid blend; writes final result to d_out.
    diffusion_step_tdm_kernel<true><<<grid, block, 0, stream>>>(
        s, u, out, alpha, beta, ax, ay, es);
}